// canODE_attentionNoValue_static_40922448396539
// MI455X (gfx1250) — compile-verified
//
#include <hip/hip_runtime.h>

typedef __attribute__((ext_vector_type(2))) float v2f;
typedef __attribute__((ext_vector_type(8))) float v8f;

#define Bsz 8
#define Nsz 2048
#define Esz 64
#define Dsz 32
#define SCALE 0.17677669529663687f /* 1/sqrt(32) */

// ---------------------------------------------------------------------------
// Kernel 1: gathered embedding -> q,k projection via V_WMMA_F32_16X16X4_F32.
// One wave per 16-row M-tile.  K (=E) = 64 in 16 steps of 4.  Four 16x16
// output tiles per wave: q[:,0:16], q[:,16:32], k[:,0:16], k[:,16:32].
//
// A (16x4 f32) per-lane layout: lane L (L&15 = M row), half = L>>4;
//   reg0 = A[M][half*2+0], reg1 = A[M][half*2+1].
// B (4x16 f32) mirrored:   reg0 = B[half*2+0][L&15], reg1 = B[half*2+1][L&15],
//   with B[e][d] = w[d*64 + e]  (w is row-major [D,E]).
// C/D (16x16 f32, 8 VGPRs): reg j holds (M = j + 8*half, Ncol = L&15).
// ---------------------------------------------------------------------------
__global__ void __launch_bounds__(256) qk_wmma_kernel(
    const int* __restrict__ pos, const float* __restrict__ val,
    const float* __restrict__ embed,
    const float* __restrict__ wq_w, const float* __restrict__ wq_b,
    const float* __restrict__ wk_w, const float* __restrict__ wk_b,
    float* __restrict__ kout, float* __restrict__ qpart) {
  const int wave = threadIdx.x >> 5;
  const int lane = threadIdx.x & 31;
  const int tile = blockIdx.x * 8 + wave;   // 0..1023 ; 16 rows each
  const int rowBase = tile * 16;
  const int kh = lane >> 4;                 // K-half (0/1)
  const int nb = lane & 15;                 // M row (for A) / N col (for B,C,D)

  const int p = pos[rowBase + nb];
  const float* eRow = embed + (long)p * Esz;
  const float* bqr0 = wq_w + nb * Esz;
  const float* bqr1 = wq_w + (nb + 16) * Esz;
  const float* bkr0 = wk_w + nb * Esz;
  const float* bkr1 = wk_w + (nb + 16) * Esz;

  v8f cq0 = {}, cq1 = {}, ck0 = {}, ck1 = {};
#pragma unroll
  for (int kk = 0; kk < 16; ++kk) {
    const int eoff = kk * 4 + kh * 2;
    v2f a   = *(const v2f*)(eRow + eoff);
    v2f bq0 = *(const v2f*)(bqr0 + eoff);
    v2f bq1 = *(const v2f*)(bqr1 + eoff);
    v2f bk0 = *(const v2f*)(bkr0 + eoff);
    v2f bk1 = *(const v2f*)(bkr1 + eoff);
    cq0 = __builtin_amdgcn_wmma_f32_16x16x4_f32(false, a, false, bq0, (short)0, cq0, false, false);
    cq1 = __builtin_amdgcn_wmma_f32_16x16x4_f32(false, a, false, bq1, (short)0, cq1, false, false);
    ck0 = __builtin_amdgcn_wmma_f32_16x16x4_f32(false, a, false, bk0, (short)0, ck0, false, false);
    ck1 = __builtin_amdgcn_wmma_f32_16x16x4_f32(false, a, false, bk1, (short)0, ck1, false, false);
  }

  const float bq0v = wq_b[nb], bq1v = wq_b[nb + 16];
  const float bk0v = wk_b[nb], bk1v = wk_b[nb + 16];

  float accq0 = 0.f, accq1 = 0.f;
#pragma unroll
  for (int j = 0; j < 8; ++j) {
    const int rr = rowBase + j + kh * 8;       // global (b,n) row of this elem
    const float vv = val[rr];
    kout[rr * Dsz + nb]      = ck0[j] + bk0v;
    kout[rr * Dsz + 16 + nb] = ck1[j] + bk1v;
    accq0 += vv * (cq0[j] + bq0v);
    accq1 += vv * (cq1[j] + bq1v);
  }
  // combine the two K-halves (same column nb, disjoint rows) deterministically
  const float o0 = accq0 + __shfl_xor(accq0, 16, 32);
  const float o1 = accq1 + __shfl_xor(accq1, 16, 32);
  if (kh == 0) {
    qpart[tile * Dsz + nb]      = o0;
    qpart[tile * Dsz + 16 + nb] = o1;
  }
}

// ---------------------------------------------------------------------------
// Kernel 2: qbar[b,d] = sum over the 128 tiles of batch b of qpart
// ---------------------------------------------------------------------------
__global__ void qbar_reduce_kernel(const float* __restrict__ qpart,
                                   float* __restrict__ qbar) {
  const int tid = threadIdx.x;     // 256 = 8 batches * 32 dims
  const int b = tid >> 5;
  const int d = tid & 31;
  float s = 0.f;
  const int tilesPerB = Nsz / 16;  // 128
#pragma unroll 8
  for (int i = 0; i < tilesPerB; ++i)
    s += qpart[(b * tilesPerB + i) * Dsz + d];
  qbar[tid] = s;
}

// ---------------------------------------------------------------------------
// Kernel 3: fx[b,j] = scale * dot(qbar[b,:], k[b,j,:])
// ---------------------------------------------------------------------------
__global__ void fx_kernel(const float* __restrict__ kmat,
                          const float* __restrict__ qbar,
                          float* __restrict__ fx) {
  const int idx = blockIdx.x * blockDim.x + threadIdx.x;  // 0..B*N-1
  const int b = idx >> 11;
  const float* kr = kmat + (long)idx * Dsz;
  const float* qb = qbar + b * Dsz;
  float s = 0.f;
#pragma unroll
  for (int d = 0; d < Dsz; ++d) s += kr[d] * qb[d];
  fx[idx] = s * SCALE;
}

// ---------------------------------------------------------------------------
// Kernel 4: RK4 replicator dynamics.  One block per batch; 256 threads x 8
// elements.  fx is constant; each stage needs one batch-wide sum(y*fx).
// ---------------------------------------------------------------------------
__device__ __forceinline__ float block_sum256(float v, float* sred) {
#pragma unroll
  for (int off = 16; off > 0; off >>= 1) v += __shfl_xor(v, off, 32);
  __syncthreads();                       // protect sred from previous round
  if ((threadIdx.x & 31) == 0) sred[threadIdx.x >> 5] = v;
  __syncthreads();
  float s = 0.f;
#pragma unroll
  for (int i = 0; i < 8; ++i) s += sred[i];
  return s;
}

__global__ void __launch_bounds__(256) ode_rk4_kernel(
    const float* __restrict__ t, const float* __restrict__ val,
    const float* __restrict__ fxg, const float* __restrict__ gate,
    float* __restrict__ out) {
  __shared__ float sred[8];
  const int b = blockIdx.x;
  const int tid = threadIdx.x;
  const float g = gate[0];

  float x[8], fx[8], k1[8], k2[8], k3[8], k4[8];
#pragma unroll
  for (int e = 0; e < 8; ++e) {
    const int n = tid + e * 256;
    x[e]  = val[b * Nsz + n];
    fx[e] = fxg[b * Nsz + n];
    out[b * Nsz + n] = x[e];              // time slice 0 = x0
  }

  for (int s = 0; s < 4; ++s) {
    const float dt = t[s + 1] - t[s];
    float p, S, y;
    // k1
    p = 0.f;
#pragma unroll
    for (int e = 0; e < 8; ++e) p += x[e] * fx[e];
    S = block_sum256(p, sred);
#pragma unroll
    for (int e = 0; e < 8; ++e) k1[e] = g * x[e] * (fx[e] - S);
    // k2
    p = 0.f;
#pragma unroll
    for (int e = 0; e < 8; ++e) { y = x[e] + 0.5f * dt * k1[e]; p += y * fx[e]; }
    S = block_sum256(p, sred);
#pragma unroll
    for (int e = 0; e < 8; ++e) { y = x[e] + 0.5f * dt * k1[e]; k2[e] = g * y * (fx[e] - S); }
    // k3
    p = 0.f;
#pragma unroll
    for (int e = 0; e < 8; ++e) { y = x[e] + 0.5f * dt * k2[e]; p += y * fx[e]; }
    S = block_sum256(p, sred);
#pragma unroll
    for (int e = 0; e < 8; ++e) { y = x[e] + 0.5f * dt * k2[e]; k3[e] = g * y * (fx[e] - S); }
    // k4
    p = 0.f;
#pragma unroll
    for (int e = 0; e < 8; ++e) { y = x[e] + dt * k3[e]; p += y * fx[e]; }
    S = block_sum256(p, sred);
#pragma unroll
    for (int e = 0; e < 8; ++e) { y = x[e] + dt * k3[e]; k4[e] = g * y * (fx[e] - S); }
    // advance + store slice s+1
    const float w = dt / 6.0f;
#pragma unroll
    for (int e = 0; e < 8; ++e) {
      x[e] += w * (k1[e] + 2.f * k2[e] + 2.f * k3[e] + k4[e]);
      out[(size_t)(s + 1) * Bsz * Nsz + b * Nsz + (tid + e * 256)] = x[e];
    }
  }
}

// ---------------------------------------------------------------------------
extern "C" void kernel_launch(void* const* d_in, const int* in_sizes, int n_in,
                              void* d_out, int out_size, void* d_ws, size_t ws_size,
                              hipStream_t stream) {
  (void)in_sizes; (void)n_in; (void)out_size; (void)ws_size;
  const float* t      = (const float*)d_in[0];
  const float* val    = (const float*)d_in[1];
  const int*   pos    = (const int*)  d_in[2];
  const float* embed  = (const float*)d_in[3];
  const float* wq_w   = (const float*)d_in[4];
  const float* wq_b   = (const float*)d_in[5];
  const float* wk_w   = (const float*)d_in[6];
  const float* wk_b   = (const float*)d_in[7];
  const float* gate   = (const float*)d_in[8];

  float* ws    = (float*)d_ws;
  float* kmat  = ws;                          // B*N*D      = 524288 floats
  float* qpart = ws + 524288;                 // 1024*32    =  32768 floats
  float* qbar  = ws + 524288 + 32768;         // B*D        =    256 floats
  float* fx    = ws + 524288 + 32768 + 256;   // B*N        =  16384 floats
  float* out   = (float*)d_out;               // T*B*N      =  81920 floats

  // 1024 tiles of 16 rows; 8 waves per block -> 128 blocks
  qk_wmma_kernel<<<128, 256, 0, stream>>>(pos, val, embed, wq_w, wq_b,
                                          wk_w, wk_b, kmat, qpart);
  qbar_reduce_kernel<<<1, 256, 0, stream>>>(qpart, qbar);
  fx_kernel<<<64, 256, 0, stream>>>(kmat, qbar, fx);
  ode_rk4_kernel<<<Bsz, 256, 0, stream>>>(t, val, fx, gate, out);
}